// Cross_Message_57363583205516
// MI455X (gfx1250) — compile-verified
//
#include <hip/hip_runtime.h>
#include <math.h>

typedef float v2f __attribute__((ext_vector_type(2)));
typedef float v8f __attribute__((ext_vector_type(8)));

#define NODE_DIM 128
#define ATTR_DIM 64
#define EPS 1e-8f

// ---------------------------------------------------------------------------
// Init: seg_max = -inf, seg_sum = 0, agg = 0
// ---------------------------------------------------------------------------
__global__ void init_kernel(float* __restrict__ seg_max,
                            float* __restrict__ seg_sum,
                            float* __restrict__ agg, int n1) {
  int i = blockIdx.x * blockDim.x + threadIdx.x;
  int total = n1 * NODE_DIM;
  if (i < total) agg[i] = 0.0f;
  if (i < n1) {
    seg_max[i] = -INFINITY;
    seg_sum[i] = 0.0f;
  }
}

// ---------------------------------------------------------------------------
// Inverse norms: one wave32 per 128-float row, float4 per lane
// ---------------------------------------------------------------------------
__global__ void inv_norm_kernel(const float* __restrict__ X,
                                float* __restrict__ inv, int n) {
  int wave = (blockIdx.x * blockDim.x + threadIdx.x) >> 5;
  int lane = threadIdx.x & 31;
  if (wave >= n) return;
  float4 a = reinterpret_cast<const float4*>(X + (size_t)wave * NODE_DIM)[lane];
  float p = a.x * a.x + a.y * a.y + a.z * a.z + a.w * a.w;
#pragma unroll
  for (int off = 16; off > 0; off >>= 1) p += __shfl_xor(p, off, 32);
  if (lane == 0) inv[wave] = 1.0f / fmaxf(sqrtf(p), EPS);
}

// ---------------------------------------------------------------------------
// Edge pass 1: cosine sim per edge (1 wave / edge) + atomic segment max
// ---------------------------------------------------------------------------
__global__ void edge_sim_kernel(const float* __restrict__ X1,
                                const float* __restrict__ X2,
                                const int* __restrict__ src,
                                const int* __restrict__ dst,
                                const float* __restrict__ invn1,
                                const float* __restrict__ invn2,
                                float* __restrict__ sim,
                                float* __restrict__ seg_max, int E) {
  int e = (blockIdx.x * blockDim.x + threadIdx.x) >> 5;
  int lane = threadIdx.x & 31;
  if (e >= E) return;
  int s = src[e];
  int d = dst[e];
  float4 a = reinterpret_cast<const float4*>(X1 + (size_t)s * NODE_DIM)[lane];
  float4 b = reinterpret_cast<const float4*>(X2 + (size_t)d * NODE_DIM)[lane];
  float p = a.x * b.x + a.y * b.y + a.z * b.z + a.w * b.w;
#pragma unroll
  for (int off = 16; off > 0; off >>= 1) p += __shfl_xor(p, off, 32);
  if (lane == 0) {
    float simv = p * invn1[s] * invn2[d];
    sim[e] = simv;
    unsafeAtomicMax(&seg_max[s], simv);  // global_atomic_max_num_f32
  }
}

// ---------------------------------------------------------------------------
// Edge pass 2: ex = exp(sim - seg_max[src]); seg_sum += ex  (1 thread / edge)
// ---------------------------------------------------------------------------
__global__ void edge_exp_kernel(const int* __restrict__ src,
                                const float* __restrict__ sim,
                                const float* __restrict__ seg_max,
                                float* __restrict__ ex,
                                float* __restrict__ seg_sum, int E) {
  int e = blockIdx.x * blockDim.x + threadIdx.x;
  if (e >= E) return;
  int s = src[e];
  float v = expf(sim[e] - seg_max[s]);
  ex[e] = v;
  unsafeAtomicAdd(&seg_sum[s], v);  // global_atomic_add_f32
}

// ---------------------------------------------------------------------------
// Edge pass 3: agg[src] += (ex/seg_sum[src]) * X2[dst]  (1 wave / edge)
// ---------------------------------------------------------------------------
__global__ void edge_agg_kernel(const float* __restrict__ X2,
                                const int* __restrict__ src,
                                const int* __restrict__ dst,
                                const float* __restrict__ ex,
                                const float* __restrict__ seg_sum,
                                float* __restrict__ agg, int E) {
  int e = (blockIdx.x * blockDim.x + threadIdx.x) >> 5;
  int lane = threadIdx.x & 31;
  if (e >= E) return;
  int s = src[e];
  int d = dst[e];
  float w = ex[e] / seg_sum[s];
  float4 b = reinterpret_cast<const float4*>(X2 + (size_t)d * NODE_DIM)[lane];
  float* out = agg + (size_t)s * NODE_DIM + lane * 4;
  unsafeAtomicAdd(out + 0, w * b.x);
  unsafeAtomicAdd(out + 1, w * b.y);
  unsafeAtomicAdd(out + 2, w * b.z);
  unsafeAtomicAdd(out + 3, w * b.w);
}

// ---------------------------------------------------------------------------
// Gate GEMM + sigmoid + elementwise multiply, via V_WMMA_F32_16X16X4_F32.
// gates = sigmoid(Xn[N1,64] @ Wg[128,64]^T); out = gates * agg
// One wave computes one 16(M) x 16(N) output tile; K=64 -> 16 WMMA ops.
//
// f32 WMMA fragment layout (ISA 7.12.2):
//   A 16x4: lanes 0-15 -> M=lane, K={0,1}; lanes 16-31 -> M=lane-16, K={2,3}
//   B 4x16: lanes 0-15 -> N=lane, K={0,1}; lanes 16-31 -> N=lane-16, K={2,3}
//   C/D 16x16: VGPR v, lane L -> M = v + (L<16?0:8), N = L%16
// ---------------------------------------------------------------------------
__global__ void gate_out_kernel(const float* __restrict__ Xn,
                                const float* __restrict__ Wg,
                                const float* __restrict__ agg,
                                float* __restrict__ out, int n1) {
  int wave = blockIdx.x * (blockDim.x >> 5) + (threadIdx.x >> 5);
  int lane = threadIdx.x & 31;
  int tilesM = (n1 + 15) >> 4;
  int tileM = wave >> 3;       // NODE_DIM/16 == 8 feature tiles
  int tileN = wave & 7;
  if (tileM >= tilesM) return;  // wave-uniform: whole wave exits before WMMA

  int l15 = lane & 15;
  int half = lane >> 4;  // 0: K pair {0,1}; 1: K pair {2,3}
  int m = tileM * 16 + l15;
  if (m >= n1) m = n1 - 1;  // clamp (N1 is a multiple of 16 anyway)
  const float* arow = Xn + (size_t)m * ATTR_DIM;
  const float* brow = Wg + (size_t)(tileN * 16 + l15) * ATTR_DIM;

  v8f c = {};
#pragma unroll
  for (int k0 = 0; k0 < ATTR_DIM; k0 += 4) {
    int k = k0 + half * 2;
    v2f a = *reinterpret_cast<const v2f*>(arow + k);
    v2f b = *reinterpret_cast<const v2f*>(brow + k);
    c = __builtin_amdgcn_wmma_f32_16x16x4_f32(
        /*neg_a=*/false, a, /*neg_b=*/false, b,
        /*c_mod=*/(short)0, c, /*reuse_a=*/false, /*reuse_b=*/false);
  }

  int n = tileN * 16 + l15;
#pragma unroll
  for (int v = 0; v < 8; ++v) {
    int row = tileM * 16 + half * 8 + v;
    if (row < n1) {
      float g = 1.0f / (1.0f + expf(-c[v]));
      size_t idx = (size_t)row * NODE_DIM + n;
      out[idx] = g * agg[idx];
    }
  }
}

// ---------------------------------------------------------------------------
// Launch
// ---------------------------------------------------------------------------
extern "C" void kernel_launch(void* const* d_in, const int* in_sizes, int n_in,
                              void* d_out, int out_size, void* d_ws,
                              size_t ws_size, hipStream_t stream) {
  const float* X_h_1 = (const float*)d_in[0];
  const float* X_h_2 = (const float*)d_in[1];
  const float* X_n_1 = (const float*)d_in[2];
  const int* cross = (const int*)d_in[3];
  const float* W_gate = (const float*)d_in[4];
  float* out = (float*)d_out;

  const int N1 = in_sizes[0] / NODE_DIM;
  const int N2 = in_sizes[1] / NODE_DIM;
  const int E = in_sizes[3] / 2;
  const int* src = cross;
  const int* dst = cross + E;

  // workspace layout (floats)
  float* ws = (float*)d_ws;
  float* invn1 = ws;               // N1
  float* invn2 = invn1 + N1;       // N2
  float* sim = invn2 + N2;         // E
  float* ex = sim + E;             // E
  float* seg_max = ex + E;         // N1
  float* seg_sum = seg_max + N1;   // N1
  float* agg = seg_sum + N1;       // N1*128

  const int BLK = 256;
  const int waves_per_blk = BLK / 32;

  // 1. init accumulators
  {
    int total = N1 * NODE_DIM;
    init_kernel<<<(total + BLK - 1) / BLK, BLK, 0, stream>>>(seg_max, seg_sum,
                                                             agg, N1);
  }
  // 2. inverse norms
  inv_norm_kernel<<<(N1 + waves_per_blk - 1) / waves_per_blk, BLK, 0, stream>>>(
      X_h_1, invn1, N1);
  inv_norm_kernel<<<(N2 + waves_per_blk - 1) / waves_per_blk, BLK, 0, stream>>>(
      X_h_2, invn2, N2);
  // 3. edge similarities + segment max
  edge_sim_kernel<<<(E + waves_per_blk - 1) / waves_per_blk, BLK, 0, stream>>>(
      X_h_1, X_h_2, src, dst, invn1, invn2, sim, seg_max, E);
  // 4. softmax numerators + segment sums
  edge_exp_kernel<<<(E + BLK - 1) / BLK, BLK, 0, stream>>>(src, sim, seg_max,
                                                           ex, seg_sum, E);
  // 5. weighted scatter-aggregate
  edge_agg_kernel<<<(E + waves_per_blk - 1) / waves_per_blk, BLK, 0, stream>>>(
      X_h_2, src, dst, ex, seg_sum, agg, E);
  // 6. gate GEMM (WMMA) + sigmoid + multiply
  {
    int tilesM = (N1 + 15) / 16;
    int waves = tilesM * (NODE_DIM / 16);
    gate_out_kernel<<<(waves + waves_per_blk - 1) / waves_per_blk, BLK, 0,
                      stream>>>(X_n_1, W_gate, agg, out, N1);
  }
}